// Back_project_50611894616890
// MI455X (gfx1250) — compile-verified
//
#include <hip/hip_runtime.h>
#include <hip/hip_bf16.h>
#include <math.h>

#define NVOX 400000
#define NV 9
#define NC 24
#define IH 120
#define IW 160

typedef __attribute__((ext_vector_type(2))) float v2f;
typedef __attribute__((ext_vector_type(8))) float v8f;

// ---------------------------------------------------------------------------
// Per-voxel projection for all 9 views.
//   mx/my  : image-space coords (may be NaN/inf if pz<=0; mask handles it)
//   wm     : weight*mask with NaN->0 (softmax input)
//   maskb  : per-view visibility bits
//   imzs   : sum of masked im_z, cnt: mask count
// ---------------------------------------------------------------------------
__device__ __forceinline__ void project_all(
    const float* __restrict__ KR, float X, float Y, float Z,
    float* mx, float* my, float* wm, unsigned& maskb, float& imzs, int& cnt)
{
    maskb = 0u; imzs = 0.f; cnt = 0;
    #pragma unroll
    for (int v = 0; v < NV; ++v) {
        const float* P = KR + v * 16;           // KRcam[v,0] row-major 4x4
        float px = P[0]*X + P[1]*Y + P[2] *Z + P[3];
        float py = P[4]*X + P[5]*Y + P[6] *Z + P[7];
        float pz = P[8]*X + P[9]*Y + P[10]*Z + P[11];
        float ix = px / pz;
        float iy = py / pz;
        float gx = 2.f * ix / (float)(IW - 1) - 1.f;
        float gy = 2.f * iy / (float)(IH - 1) - 1.f;
        // NaN compares are false -> mask false, matching the reference
        bool m = (fabsf(gx) <= 1.f) && (fabsf(gy) <= 1.f) && (pz > 0.f);
        float wgt = 1.f / (sqrtf(ix*ix + iy*iy) + 1e-7f);
        float w = wgt * (m ? 1.f : 0.f);
        if (w != w) w = 0.f;                    // NaN -> 0 (reference semantics)
        mx[v] = ix; my[v] = iy; wm[v] = w;
        if (m) { maskb |= 1u << v; imzs += pz; cnt++; }
    }
}

__device__ __forceinline__ void softmax9(float* w)
{
    float m = w[0];
    #pragma unroll
    for (int i = 1; i < NV; ++i) m = fmaxf(m, w[i]);
    float s = 0.f;
    #pragma unroll
    for (int i = 0; i < NV; ++i) { w[i] = expf(w[i] - m); s += w[i]; }
    float r = 1.f / s;
    #pragma unroll
    for (int i = 0; i < NV; ++i) w[i] *= r;
}

// ---------------------------------------------------------------------------
// Bilinear-sample all channels of each visible view, accumulate with weight
// wv[v] into outc[0..23].  Only mask-true views touched (masked-out views
// contribute exactly 0 in the reference after features*mask).
// ---------------------------------------------------------------------------
__device__ __forceinline__ void sample_accum(
    const float* __restrict__ feats, unsigned maskb,
    const float* mx, const float* my, const float* wv, float* outc)
{
    for (int v = 0; v < NV; ++v) {
        if (!((maskb >> v) & 1u)) continue;
        float wvv = wv[v];
        float ix = mx[v], iy = my[v];          // mask true => ix in [0,159], iy in [0,119]
        float x0f = floorf(ix), y0f = floorf(iy);
        float wx1 = ix - x0f, wy1 = iy - y0f;
        float wx0 = 1.f - wx1, wy0 = 1.f - wy1;
        int x0 = (int)x0f, y0 = (int)y0f;
        bool vx1 = (x0 + 1) <= (IW - 1);
        bool vy1 = (y0 + 1) <= (IH - 1);
        int x1c = vx1 ? x0 + 1 : IW - 1;
        int y1c = vy1 ? y0 + 1 : IH - 1;
        float w00 = wx0 * wy0;
        float w10 = vx1 ? wx1 * wy0 : 0.f;
        float w01 = vy1 ? wx0 * wy1 : 0.f;
        float w11 = (vx1 && vy1) ? wx1 * wy1 : 0.f;
        const float* base = feats + (size_t)v * NC * IH * IW;
        int o00 = y0  * IW + x0;
        int o10 = y0  * IW + x1c;
        int o01 = y1c * IW + x0;
        int o11 = y1c * IW + x1c;
        #pragma unroll
        for (int c = 0; c < NC; ++c) {
            const float* pc = base + c * (IH * IW);
            float val = pc[o00]*w00 + pc[o10]*w10 + pc[o01]*w01 + pc[o11]*w11;
            outc[c] += val * wvv;
        }
    }
}

// ---------------------------------------------------------------------------
// Kernel 1: global statistics.
//   lw = fsum @ w_lin + b_lin computed with V_WMMA_F32_16X16X4_F32
//   (16-voxel tiles staged through LDS, stride 25 to avoid bank conflicts).
//   Accumulates per-view sum(lw), sum(lw^2) and depth pos/sum/sumsq into
//   f64 global accumulators (block-local f32 LDS atomics first).
// ---------------------------------------------------------------------------
__global__ __launch_bounds__(256) void bp_stats(
    const int*   __restrict__ coords, const float* __restrict__ origin,
    const float* __restrict__ vsz,    const float* __restrict__ feats,
    const float* __restrict__ KR,     const float* __restrict__ w_lin,
    const float* __restrict__ b_lin,  double* __restrict__ acc)
{
    __shared__ float fsumS[256 * 25];
    __shared__ float sAcc[24];
    const int tid = threadIdx.x;
    if (tid < 24) sAcc[tid] = 0.f;
    __syncthreads();

    const int n = blockIdx.x * 256 + tid;
    float fsum[NC];
    #pragma unroll
    for (int c = 0; c < NC; ++c) fsum[c] = 0.f;

    if (n < NVOX) {
        const float vs = vsz[0];
        float X = (float)coords[n*4 + 1] * vs + origin[0];
        float Y = (float)coords[n*4 + 2] * vs + origin[1];
        float Z = (float)coords[n*4 + 3] * vs + origin[2];
        float mx[NV], my[NV], wm[NV]; unsigned mb; float imzs; int cnt;
        project_all(KR, X, Y, Z, mx, my, wm, mb, imzs, cnt);
        softmax9(wm);
        sample_accum(feats, mb, mx, my, wm, fsum);
        float imz = imzs / (float)(cnt > 0 ? cnt : 1);
        if (imz > 0.f) {
            atomicAdd(&sAcc[18], 1.f);
            atomicAdd(&sAcc[19], imz);
            atomicAdd(&sAcc[20], imz * imz);
        }
    }
    #pragma unroll
    for (int c = 0; c < NC; ++c) fsumS[tid * 25 + c] = fsum[c];
    __syncthreads();

    // ---- WMMA: lw[16 voxels][9 views] per tile, 6 K-steps of 4 over C=24 ----
    const int wave = tid >> 5, lane = tid & 31;
    const int half = lane >> 4, l16 = lane & 15;

    float b0[6], b1[6];                        // B tile: w_lin (K=C x N=views)
    #pragma unroll
    for (int kk = 0; kk < 6; ++kk) {
        int k0 = kk * 4 + half * 2;
        b0[kk] = (l16 < NV) ? w_lin[(k0    ) * NV + l16] : 0.f;
        b1[kk] = (l16 < NV) ? w_lin[(k0 + 1) * NV + l16] : 0.f;
    }
    #pragma unroll
    for (int tt = 0; tt < 2; ++tt) {           // 16 tiles / 8 waves
        int t = wave + tt * 8;                 // uniform per wave: EXEC all-1s
        v8f acc8 = {0.f, 0.f, 0.f, 0.f, 0.f, 0.f, 0.f, 0.f};
        #pragma unroll
        for (int kk = 0; kk < 6; ++kk) {
            int k0 = kk * 4 + half * 2;        // A: VGPR0=K0|K2, VGPR1=K1|K3
            v2f A, B;
            A[0] = fsumS[(t * 16 + l16) * 25 + k0];
            A[1] = fsumS[(t * 16 + l16) * 25 + k0 + 1];
            B[0] = b0[kk];
            B[1] = b1[kk];
            acc8 = __builtin_amdgcn_wmma_f32_16x16x4_f32(
                false, A, false, B, (short)0, acc8, false, false);
        }
        if (l16 < NV) {                        // D: M = r + 8*half, N = l16
            float bl = b_lin[l16];
            float s = 0.f, s2 = 0.f;
            #pragma unroll
            for (int r = 0; r < 8; ++r) {
                int vox = blockIdx.x * 256 + t * 16 + half * 8 + r;
                if (vox < NVOX) {
                    float lw = acc8[r] + bl;
                    s += lw; s2 += lw * lw;
                }
            }
            atomicAdd(&sAcc[l16], s);
            atomicAdd(&sAcc[9 + l16], s2);
        }
    }
    __syncthreads();
    if (tid < 21) atomicAdd(&acc[tid], (double)sAcc[tid]);
}

// ---------------------------------------------------------------------------
// Kernel 2: finalize batchnorm + depth statistics (1 block, 32 threads).
//   fin[0..8]=mu, fin[9..17]=gamma/sqrt(var+eps), fin[18]=depth mean,
//   fin[19]=1/(sqrt(sum sq dev)+1e-5)
// ---------------------------------------------------------------------------
__global__ void bp_finalize(const double* __restrict__ acc,
                            float* __restrict__ fin,
                            const float* __restrict__ gamma)
{
    int t = threadIdx.x;
    if (t < NV) {
        double mu  = acc[t] / (double)NVOX;
        double var = acc[9 + t] / (double)NVOX - mu * mu;
        if (var < 0.0) var = 0.0;
        fin[t]     = (float)mu;
        fin[9 + t] = (float)((double)gamma[t] / sqrt(var + 1e-5));
    }
    if (t == 0) {
        double cp = acc[18], s = acc[19], s2 = acc[20];
        double npos = (cp > 0.0) ? cp : 1.0;
        double mean = s / npos;
        double ssd  = s2 - 2.0 * mean * s + cp * mean * mean;
        if (ssd < 0.0) ssd = 0.0;
        fin[18] = (float)mean;
        fin[19] = (float)(1.0 / (sqrt(ssd) + 1e-5));
    }
}

// ---------------------------------------------------------------------------
// Kernel 3: final outputs.  Recomputes projection/sampling (feats are
// L2-resident: 16.6 MB vs 192 MB L2), applies BN+relu+mask+softmax,
// re-samples with the new weights, writes (N,25) features + count.
// ---------------------------------------------------------------------------
__global__ __launch_bounds__(256) void bp_out(
    const int*   __restrict__ coords, const float* __restrict__ origin,
    const float* __restrict__ vsz,    const float* __restrict__ feats,
    const float* __restrict__ KR,     const float* __restrict__ w_lin,
    const float* __restrict__ b_lin,  const float* __restrict__ bn_beta,
    const float* __restrict__ fin,    float* __restrict__ out)
{
    __shared__ float sW[NC * NV];
    __shared__ float sB[NV], sBeta[NV], sMu[NV], sScale[NV];
    const int tid = threadIdx.x;
    if (tid < NC * NV) sW[tid] = w_lin[tid];
    if (tid >= 224 && tid < 224 + NV) {
        int v = tid - 224;
        sB[v]     = b_lin[v];
        sBeta[v]  = bn_beta[v];
        sMu[v]    = fin[v];
        sScale[v] = fin[9 + v];
    }
    __syncthreads();
    const float meanZ  = fin[18];
    const float invStd = fin[19];

    const int n = blockIdx.x * 256 + tid;
    if (n >= NVOX) return;

    const float vs = vsz[0];
    float X = (float)coords[n*4 + 1] * vs + origin[0];
    float Y = (float)coords[n*4 + 2] * vs + origin[1];
    float Z = (float)coords[n*4 + 3] * vs + origin[2];
    float mx[NV], my[NV], wm[NV]; unsigned mb; float imzs; int cnt;
    project_all(KR, X, Y, Z, mx, my, wm, mb, imzs, cnt);
    softmax9(wm);

    float fsum[NC];
    #pragma unroll
    for (int c = 0; c < NC; ++c) fsum[c] = 0.f;
    sample_accum(feats, mb, mx, my, wm, fsum);

    // linear -> BN -> relu -> mask -> softmax  (all finite by construction)
    float h[NV];
    #pragma unroll
    for (int j = 0; j < NV; ++j) {
        float lw = sB[j];
        #pragma unroll
        for (int c = 0; c < NC; ++c) lw += fsum[c] * sW[c * NV + j];
        float t = sScale[j] * (lw - sMu[j]) + sBeta[j];
        t = fmaxf(t, 0.f);
        if (!((mb >> j) & 1u)) t = 0.f;
        h[j] = t;
    }
    softmax9(h);

    float feat[NC];
    #pragma unroll
    for (int c = 0; c < NC; ++c) feat[c] = 0.f;
    sample_accum(feats, mb, mx, my, h, feat);

    size_t rb = (size_t)n * 25;
    #pragma unroll
    for (int c = 0; c < NC; ++c) out[rb + c] = feat[c];
    float imz = imzs / (float)(cnt > 0 ? cnt : 1);
    out[rb + 24] = (imz > 0.f) ? (imz - meanZ) * invStd : 0.f;
    out[(size_t)NVOX * 25 + n] = (float)cnt;
}

// ---------------------------------------------------------------------------
extern "C" void kernel_launch(void* const* d_in, const int* in_sizes, int n_in,
                              void* d_out, int out_size, void* d_ws, size_t ws_size,
                              hipStream_t stream)
{
    const int*   coords = (const int*)  d_in[0];
    const float* origin = (const float*)d_in[1];
    const float* vsz    = (const float*)d_in[2];
    const float* feats  = (const float*)d_in[3];
    const float* KR     = (const float*)d_in[4];
    const float* w_lin  = (const float*)d_in[5];
    const float* b_lin  = (const float*)d_in[6];
    const float* gamma  = (const float*)d_in[7];
    const float* beta   = (const float*)d_in[8];
    float*  out = (float*)d_out;
    double* acc = (double*)d_ws;                       // 21 f64 accumulators
    float*  fin = (float*)((char*)d_ws + 256);         // 20 finalized floats

    (void)in_sizes; (void)n_in; (void)out_size; (void)ws_size;

    hipMemsetAsync(d_ws, 0, 512, stream);              // deterministic per call

    dim3 grid((NVOX + 255) / 256);
    bp_stats<<<grid, 256, 0, stream>>>(coords, origin, vsz, feats, KR,
                                       w_lin, b_lin, acc);
    bp_finalize<<<1, 32, 0, stream>>>(acc, fin, gamma);
    bp_out<<<grid, 256, 0, stream>>>(coords, origin, vsz, feats, KR,
                                     w_lin, b_lin, beta, fin, out);
}